// FeatureVectorInteractions_88046829568431
// MI455X (gfx1250) — compile-verified
//
#include <hip/hip_runtime.h>

typedef __attribute__((ext_vector_type(2))) float v2f;
typedef __attribute__((ext_vector_type(8))) float v8f;

// One block per batch. N=128, D=64 fixed by the reference.
// 256 threads = 8 wave32 waves.
__global__ __launch_bounds__(256) void FeatureVectorInteractions_kernel(
    const int* __restrict__ VI0, const int* __restrict__ VI1,
    const float* __restrict__ V0, const float* __restrict__ V1,
    float* __restrict__ out)
{
    __shared__ float sm0[128];
    __shared__ float sm1[128];
    __shared__ float s0[16 * 64];   // [g][d] partial masked sums of V0
    __shared__ float s1[16 * 64];   // [g][d] partial masked sums of V1
    __shared__ float pbuf[64];      // p[d] = u0[d] * u1[d]

    const int b = blockIdx.x;
    const int t = threadIdx.x;

    // Stage 0/1 masks into LDS (clip(vi, 0, 1)).
    if (t < 128) {
        int v = VI0[b * 128 + t];
        sm0[t] = (float)(v < 0 ? 0 : (v > 1 ? 1 : v));
        v = VI1[b * 128 + t];
        sm1[t] = (float)(v < 0 ? 0 : (v > 1 ? 1 : v));
    }
    __syncthreads();

    // Streaming masked accumulation. Flat float4 index f = k*256 + t:
    //   row  i  = f / 16 = k*16 + (t>>4)   (16 float4 per 64-float row)
    //   dblk    = f % 16 = t & 15          (constant per thread)
    const float4* v0p = (const float4*)V0 + (size_t)b * 2048;  // 128*64/4
    const float4* v1p = (const float4*)V1 + (size_t)b * 2048;
    const int d4 = t & 15;
    const int g  = t >> 4;

    float a0x = 0.f, a0y = 0.f, a0z = 0.f, a0w = 0.f;
    float a1x = 0.f, a1y = 0.f, a1z = 0.f, a1w = 0.f;
#pragma unroll
    for (int k = 0; k < 8; ++k) {
        const int i = k * 16 + g;
        const int f = i * 16 + d4;          // == k*256 + t, coalesced b128 loads
        const float m0v = sm0[i];
        const float4 x = v0p[f];
        a0x += m0v * x.x; a0y += m0v * x.y; a0z += m0v * x.z; a0w += m0v * x.w;
        const float m1v = sm1[i];
        const float4 y = v1p[f];
        a1x += m1v * y.x; a1y += m1v * y.y; a1z += m1v * y.z; a1w += m1v * y.w;
    }

    // s0[g][4*d4 + c] laid out so s0[g*64 + d] is the g-th partial of u0[d].
    s0[4 * t + 0] = a0x; s0[4 * t + 1] = a0y; s0[4 * t + 2] = a0z; s0[4 * t + 3] = a0w;
    s1[4 * t + 0] = a1x; s1[4 * t + 1] = a1y; s1[4 * t + 2] = a1z; s1[4 * t + 3] = a1w;
    __syncthreads();

    // Fold 16 partials per d; form p[d] = u0[d]*u1[d]. Bank-conflict free.
    if (t < 64) {
        float u0 = 0.f, u1 = 0.f;
#pragma unroll
        for (int gg = 0; gg < 16; ++gg) {
            u0 += s0[gg * 64 + t];
            u1 += s1[gg * 64 + t];
        }
        pbuf[t] = u0 * u1;
    }
    __syncthreads();

    // Final 64-wide dot reduction via one V_WMMA_F32_16X16X4_F32.
    // A = ones(16x4)  =>  D[m,n] = sum_k B[k,n]   (column sums).
    // Sum over row 0 of D == sum of ALL 64 B elements, independent of the
    // slot->(k,n) mapping, so stuffing pbuf[lane] / pbuf[lane+32] into B's
    // two VGPRs is exact regardless of B's striping order.
    // Executed by every wave (EXEC all 1s, per WMMA restriction); redundant
    // but free. Only thread 0 stores.
    const int lane = t & 31;
    v2f amat; amat[0] = 1.0f; amat[1] = 1.0f;
    v2f bmat; bmat[0] = pbuf[lane]; bmat[1] = pbuf[lane + 32];
    v8f cacc = {};
    cacc = __builtin_amdgcn_wmma_f32_16x16x4_f32(
        /*neg_a=*/false, amat, /*neg_b=*/false, bmat,
        /*c_mod=*/(short)0, cacc, /*reuse_a=*/false, /*reuse_b=*/false);

    // Row 0 of D: VGPR0 of lanes 0..15 (lanes 16..31 hold row 8 -> masked out).
    float val = (lane < 16) ? cacc[0] : 0.0f;
    val += __shfl_xor(val, 8, 32);
    val += __shfl_xor(val, 4, 32);
    val += __shfl_xor(val, 2, 32);
    val += __shfl_xor(val, 1, 32);

    if (t == 0) out[b] = val;
}

extern "C" void kernel_launch(void* const* d_in, const int* in_sizes, int n_in,
                              void* d_out, int out_size, void* d_ws, size_t ws_size,
                              hipStream_t stream) {
    const int*   VI0 = (const int*)d_in[0];
    const int*   VI1 = (const int*)d_in[1];
    const float* V0  = (const float*)d_in[2];
    const float* V1  = (const float*)d_in[3];
    float*       out = (float*)d_out;

    const int B = in_sizes[0] / 128;   // N = 128
    FeatureVectorInteractions_kernel<<<B, 256, 0, stream>>>(VI0, VI1, V0, V1, out);
}